// FocalLoss_20409684591021
// MI455X (gfx1250) — compile-verified
//
#include <hip/hip_runtime.h>
#include <math.h>

typedef __attribute__((ext_vector_type(2))) float v2f;
typedef __attribute__((ext_vector_type(8))) float v8f;

#define NBLOCKS  512
#define NTHREADS 256

// DPP8 selector packing: lane i reads lane (rowbase + a_i), 3 bits each.
#define DPP8SEL(a0,a1,a2,a3,a4,a5,a6,a7) \
  ((a0)|((a1)<<3)|((a2)<<6)|((a3)<<9)|((a4)<<12)|((a5)<<15)|((a6)<<18)|((a7)<<21))
#define SEL_XOR1 DPP8SEL(1,0,3,2,5,4,7,6)
#define SEL_XOR2 DPP8SEL(2,3,0,1,6,7,4,5)
#define SEL_XOR4 DPP8SEL(4,5,6,7,0,1,2,3)

#define DPPADD(s, SEL) \
  (s) += __int_as_float(__builtin_amdgcn_mov_dpp8(__float_as_int(s), (SEL)))

// Fixed mixing matrix M [3][8] and bias [8] from the reference.
__constant__ float c_M[3][8] = {
    {1.0f, 0.5f, 0.5f, 0.0f, 0.0f, 1.0f/3.0f, 0.0f, -1.0f/3.0f},
    {0.0f, 0.0f, 0.5f, 0.0f, 1.0f, 1.0f/3.0f, 0.5f, -1.0f/3.0f},
    {0.0f, 0.5f, 0.0f, 1.0f, 0.0f, 1.0f/3.0f, 0.5f, -1.0f/3.0f},
};
__constant__ float c_Bias[8] = {0.f, 0.f, 0.f, 0.f, 0.f, 0.f, 0.f, 1.0f};

__device__ __forceinline__ float focal_from_logits8(const float* l, int tgt) {
    // logits are tiny (inputs in [0,1)) -> no max subtraction needed
    float s = 0.f;
    #pragma unroll
    for (int j = 0; j < 8; ++j) s += __expf(l[j]);
    float logp = __logf(s) - l[tgt];   // CE loss
    float p    = __expf(-logp);
    float om   = 1.0f - p;
    return om * om * logp;             // GAMMA == 2
}

__global__ __launch_bounds__(NTHREADS) void focal_wmma_kernel(
    const float* __restrict__ x, const int* __restrict__ target,
    float* __restrict__ partial, int n)
{
    const int lane         = threadIdx.x & 31;
    const int waveInBlock  = threadIdx.x >> 5;
    const int wavesPerBlk  = NTHREADS / 32;
    const int globalWave   = blockIdx.x * wavesPerBlk + waveInBlock;
    const int totalWaves   = gridDim.x * wavesPerBlk;
    const int numTiles     = n >> 4;           // 16 rows per WMMA tile

    const int  col  = lane & 15;               // B: column / A: row-in-tile
    const bool hi   = lane >= 16;
    const int  g8   = lane & 7;                // position within 8-lane group
    const bool validGrp = (lane & 8) == 0;     // groups holding classes 0..7
    const int  tOff = g8 + (hi ? 8 : 0);       // row whose target this lane caches

    // ---- Constant B (4x16, K x N) per ISA layout:
    //   VGPR0: lanes0-15 K=0, lanes16-31 K=2 ; VGPR1: lanes0-15 K=1, lanes16-31 K=3
    // K=3 row is the bias (A's K=3 column is ones). Columns 8..15 = 0.
    v2f bmat;
    bmat.x = (col < 8) ? c_M[hi ? 2 : 0][col] : 0.0f;
    bmat.y = (col < 8) ? (hi ? c_Bias[col] : c_M[1][col]) : 0.0f;

    float acc = 0.0f;

    for (int tile = globalWave; tile < numTiles; tile += totalWaves) {
        const int rowBase = tile << 4;
        const int row     = rowBase + col;

        // ---- A (16x4): lanes0-15 -> {x0, x1}; lanes16-31 -> {x2, 1.0}
        v2f a;
        if (!hi) { a.x = x[row * 3 + 0]; a.y = x[row * 3 + 1]; }
        else     { a.x = x[row * 3 + 2]; a.y = 1.0f; }

        // One target per lane; per-component DPP8 broadcast distributes them.
        const int tloc = target[rowBase + tOff];

        // Prefetch next grid-stride tile (global_prefetch_b8 on gfx1250).
        if (tile + totalWaves < numTiles)
            __builtin_prefetch(&x[(row + totalWaves * 16) * 3], 0, 0);

        v8f d = {};
        // D = A(16x4) x B(4x16) + 0 : logits for 16 rows (cols 8..15 = 0)
        d = __builtin_amdgcn_wmma_f32_16x16x4_f32(
                false, a, false, bmat, (short)0, d, false, false);

        // D layout: component r = rows r / r+8 (lane halves), class = lane&15.
        // All cross-lane ops stay inside 8-lane DPP8 rows -> pure VALU.
        #define PROC_COMPONENT(r) do {                                         \
            float logit = d[r];                                                \
            float e = __expf(logit);          /* logits in [-1/3, 1] */        \
            float s = e;                                                       \
            DPPADD(s, SEL_XOR1);                                               \
            DPPADD(s, SEL_XOR2);                                               \
            DPPADD(s, SEL_XOR4);                                               \
            int tb = __builtin_amdgcn_mov_dpp8(                                \
                         tloc, DPP8SEL(r,r,r,r,r,r,r,r));                      \
            float logp = __logf(s) - logit;   /* CE at this class */           \
            float p    = e * __builtin_amdgcn_rcpf(s);                         \
            float om   = 1.0f - p;                                             \
            float contrib = om * om * logp;                                    \
            bool  mine = validGrp && (g8 == tb);                               \
            acc += mine ? contrib : 0.0f;                                      \
        } while (0)

        PROC_COMPONENT(0); PROC_COMPONENT(1);
        PROC_COMPONENT(2); PROC_COMPONENT(3);
        PROC_COMPONENT(4); PROC_COMPONENT(5);
        PROC_COMPONENT(6); PROC_COMPONENT(7);
        #undef PROC_COMPONENT
    }

    // ---- scalar tail (n not multiple of 16), handled by wave 0
    const int tailStart = numTiles << 4;
    if (globalWave == 0 && (tailStart + lane) < n) {
        const int r = tailStart + lane;
        float x0 = x[r * 3 + 0], x1 = x[r * 3 + 1], x2 = x[r * 3 + 2];
        float l[8];
        float t3 = (x0 + x1 + x2) * (1.0f / 3.0f);
        l[0] = x0;               l[1] = 0.5f * (x0 + x2);
        l[2] = 0.5f * (x0 + x1); l[3] = x2;
        l[4] = x1;               l[5] = t3;
        l[6] = 0.5f * (x1 + x2); l[7] = 1.0f - t3;
        acc += focal_from_logits8(l, target[r]);
    }

    // ---- wave32 reduce (once per wave; bpermute cost negligible here)
    #pragma unroll
    for (int m = 16; m >= 1; m >>= 1) acc += __shfl_xor(acc, m, 32);

    __shared__ float lds[NTHREADS / 32];
    if (lane == 0) lds[waveInBlock] = acc;
    __syncthreads();
    if (threadIdx.x == 0) {
        float s = 0.f;
        #pragma unroll
        for (int i = 0; i < NTHREADS / 32; ++i) s += lds[i];
        partial[blockIdx.x] = s;
    }
}

__global__ __launch_bounds__(256) void focal_final_reduce(
    const float* __restrict__ partial, float* __restrict__ out, float invN)
{
    float v = partial[threadIdx.x] + partial[threadIdx.x + 256];
    #pragma unroll
    for (int m = 16; m >= 1; m >>= 1) v += __shfl_xor(v, m, 32);

    __shared__ float lds[8];
    const int lane = threadIdx.x & 31, w = threadIdx.x >> 5;
    if (lane == 0) lds[w] = v;
    __syncthreads();
    if (threadIdx.x == 0) {
        float s = 0.f;
        #pragma unroll
        for (int i = 0; i < 8; ++i) s += lds[i];
        out[0] = s * invN;
    }
}

extern "C" void kernel_launch(void* const* d_in, const int* in_sizes, int n_in,
                              void* d_out, int out_size, void* d_ws, size_t ws_size,
                              hipStream_t stream) {
    const float* x   = (const float*)d_in[0];
    const int*   tgt = (const int*)d_in[1];
    float*       out = (float*)d_out;
    float*       partial = (float*)d_ws;     // NBLOCKS floats
    const int n = in_sizes[0] / 3;

    focal_wmma_kernel<<<NBLOCKS, NTHREADS, 0, stream>>>(x, tgt, partial, n);
    focal_final_reduce<<<1, 256, 0, stream>>>(partial, out, 1.0f / (float)n);
}